// Features_52329881534748
// MI455X (gfx1250) — compile-verified
//
#include <hip/hip_runtime.h>
#include <stdint.h>

#define BN    4
#define NPTS  8192
#define HH    2048
#define WW    2048
#define RAD   4
#define KSEL  1024
#define TILE  256
#define NTILES (NPTS / TILE)
#define SORT_THREADS 1024

#define AS1 __attribute__((address_space(1)))
#define AS3 __attribute__((address_space(3)))

typedef int v2i __attribute__((ext_vector_type(2)));

#if defined(__AMDGCN__) && __has_builtin(__builtin_amdgcn_global_load_async_to_lds_b64)
#define USE_ASYNC64 1
#define ISSUES_PER_TILE 1
#elif defined(__AMDGCN__) && __has_builtin(__builtin_amdgcn_global_load_async_to_lds_b32)
#define USE_ASYNC32 1
#define ISSUES_PER_TILE 2
#endif

#if defined(USE_ASYNC64) || defined(USE_ASYNC32)
#define USE_ASYNC_LDS 1
#if __has_builtin(__builtin_amdgcn_s_wait_asynccnt)
#define WAIT_ASYNC(n) __builtin_amdgcn_s_wait_asynccnt(n)
#else
#define WAIT_ASYNC(n) asm volatile("s_wait_asynccnt " #n ::: "memory")
#endif
#endif

__device__ __forceinline__ float neg_inf_f() { return __int_as_float((int)0xFF800000u); }

// ---------------------------------------------------------------------------
// Kernel 0: round (half-even), clip, pack pixel coords x|y<<16 and score bits
// into one uint2 per keypoint (8-byte records -> single b64 async tile loads).
// ---------------------------------------------------------------------------
__global__ void __launch_bounds__(256)
prep_pack(const float* __restrict__ lafs, const float* __restrict__ scores,
          uint2* __restrict__ kp) {
  int g = blockIdx.x * blockDim.x + threadIdx.x;
  if (g >= BN * NPTS) return;
  float xf = lafs[(size_t)g * 6 + 2];   // lafs[b,i,0,2] = x
  float yf = lafs[(size_t)g * 6 + 5];   // lafs[b,i,1,2] = y
  int x = (int)rintf(xf); x = min(max(x, 0), WW - 1);
  int y = (int)rintf(yf); y = min(max(y, 0), HH - 1);
  uint2 r;
  r.x = (uint32_t)x | ((uint32_t)y << 16);
  r.y = __float_as_uint(scores[g]);
  kp[g] = r;
}

// ---------------------------------------------------------------------------
// Kernel 1: pairwise windowed-max NMS.
// survive(i) <=> score_i == max over j with |dx|<=4 && |dy|<=4 (includes j==i,
// which reproduces the scatter-max per-pixel dedup of the reference).
// j-tiles DMAed into double-buffered LDS with gfx1250 async copies; tile t+1
// is in flight while tile t is processed (partial s_wait_asynccnt).
// ---------------------------------------------------------------------------
__device__ __forceinline__ void issue_tile(const uint2* kb, int base, uint2* dst) {
#if defined(USE_ASYNC64)
  __builtin_amdgcn_global_load_async_to_lds_b64(
      (AS1 v2i*)(void*)(kb + base + threadIdx.x),
      (AS3 v2i*)(void*)dst, 0, 0);
#elif defined(USE_ASYNC32)
  __builtin_amdgcn_global_load_async_to_lds_b32(
      (AS1 int*)(void*)((const uint32_t*)(kb + base + threadIdx.x)),
      (AS3 int*)(void*)dst, 0, 0);
  __builtin_amdgcn_global_load_async_to_lds_b32(
      (AS1 int*)(void*)(((const uint32_t*)(kb + base + threadIdx.x)) + 1),
      (AS3 int*)(void*)(((uint32_t*)dst) + 1), 0, 0);
#else
  *dst = kb[base + threadIdx.x];
#endif
}

__global__ void __launch_bounds__(TILE)
nms_mask(const uint2* __restrict__ kp, float* __restrict__ masked) {
  const int blocksPerBatch = NPTS / TILE;
  const int b = blockIdx.x / blocksPerBatch;
  const int i = (blockIdx.x % blocksPerBatch) * TILE + threadIdx.x;

  const uint2* kb = kp + (size_t)b * NPTS;

  const uint2 me = kb[i];
  const int xi = (int)(me.x & 0xFFFFu);
  const int yi = (int)(me.x >> 16);
  const float si = __uint_as_float(me.y);

  float m = neg_inf_f();

  __shared__ uint2 buf[2][TILE];

  issue_tile(kb, 0, &buf[0][threadIdx.x]);

  for (int t = 0; t < NTILES; ++t) {
    const int cur = t & 1;
    if (t + 1 < NTILES)
      issue_tile(kb, (t + 1) * TILE, &buf[cur ^ 1][threadIdx.x]);
#ifdef USE_ASYNC_LDS
    if (t + 1 < NTILES) WAIT_ASYNC(ISSUES_PER_TILE);  // tile t landed, t+1 may fly
    else                WAIT_ASYNC(0);
#endif
    __syncthreads();   // all waves' DMA for tile t visible

#pragma unroll 16
    for (int u = 0; u < TILE; ++u) {
      const uint2 e = buf[cur][u];
      const int dx = abs(xi - (int)(e.x & 0xFFFFu));
      const int dy = abs(yi - (int)(e.x >> 16));
      const float sj = __uint_as_float(e.y);
      const bool nearby = (dx <= RAD) && (dy <= RAD);
      m = (nearby && sj > m) ? sj : m;
    }
    __syncthreads();   // tile t fully consumed before buffer reuse at t+2
  }

  masked[(size_t)b * NPTS + i] = (si == m) ? si : neg_inf_f();
}

// ---------------------------------------------------------------------------
// Kernel 2: per-batch full bitonic sort of 8192 (score desc, index asc) keys
// in 64KB LDS; emit top-1024 scores + gathered 2x3 LAFs.
// key = (~orderable(score) << 32) | index, sorted ascending -> best first.
// ---------------------------------------------------------------------------
__global__ void __launch_bounds__(SORT_THREADS)
topk_sort(const float* __restrict__ lafs, const float* __restrict__ masked,
          float* __restrict__ out) {
  __shared__ unsigned long long keys[NPTS];   // 64 KB
  const int b = blockIdx.x;
  const int tid = threadIdx.x;
  const float* mb = masked + (size_t)b * NPTS;

  for (int i = tid; i < NPTS; i += SORT_THREADS) {
    uint32_t bits = __float_as_uint(mb[i]);
    uint32_t u = ((int)bits < 0) ? ~bits : (bits | 0x80000000u);  // monotone map
    keys[i] = ((unsigned long long)(~u) << 32) | (uint32_t)i;
  }

  for (unsigned kk = 2; kk <= NPTS; kk <<= 1) {
    for (unsigned j = kk >> 1; j > 0; j >>= 1) {
      __syncthreads();
      for (unsigned i = tid; i < NPTS; i += SORT_THREADS) {
        const unsigned l = i ^ j;
        if (l > i) {
          const unsigned long long a = keys[i];
          const unsigned long long c = keys[l];
          const bool up = ((i & kk) == 0);
          const bool sw = up ? (a > c) : (a < c);
          if (sw) { keys[i] = c; keys[l] = a; }
        }
      }
    }
  }
  __syncthreads();

  const float* lb = lafs + (size_t)b * NPTS * 6;
  float* lafs_out   = out;                          // [B,K,2,3]
  float* scores_out = out + (size_t)BN * KSEL * 6;  // [B,K]

  for (int t = tid; t < KSEL; t += SORT_THREADS) {
    const unsigned long long key = keys[t];
    const uint32_t idx = (uint32_t)(key & 0xFFFFFFFFu);
    const uint32_t u   = ~((uint32_t)(key >> 32));
    const uint32_t bits = (u & 0x80000000u) ? (u ^ 0x80000000u) : ~u;  // inverse map
    scores_out[(size_t)b * KSEL + t] = __uint_as_float(bits);
    const size_t ob = ((size_t)b * KSEL + t) * 6;
    const size_t ib = (size_t)idx * 6;
#pragma unroll
    for (int e = 0; e < 6; ++e) lafs_out[ob + e] = lb[ib + e];
  }
}

// ---------------------------------------------------------------------------
extern "C" void kernel_launch(void* const* d_in, const int* in_sizes, int n_in,
                              void* d_out, int out_size, void* d_ws, size_t ws_size,
                              hipStream_t stream) {
  const float* lafs   = (const float*)d_in[0];   // [B,N,2,3] f32
  const float* scores = (const float*)d_in[1];   // [B,N] f32
  // d_in[2]=h, d_in[3]=w : compile-time constants (2048), ignored.

  uint2* kp     = (uint2*)d_ws;                                        // B*N uint2
  float* masked = (float*)((char*)d_ws + (size_t)BN * NPTS * 8);       // B*N f32
  float* out    = (float*)d_out;

  hipLaunchKernelGGL(prep_pack, dim3((BN * NPTS + 255) / 256), dim3(256), 0, stream,
                     lafs, scores, kp);
  hipLaunchKernelGGL(nms_mask, dim3(BN * (NPTS / TILE)), dim3(TILE), 0, stream,
                     kp, masked);
  hipLaunchKernelGGL(topk_sort, dim3(BN), dim3(SORT_THREADS), 0, stream,
                     lafs, masked, out);
}